// RelationAwareAttention_10703058502268
// MI455X (gfx1250) — compile-verified
//
#include <hip/hip_runtime.h>
#include <hip/hip_bf16.h>

#define BSZ 4
#define NH 8
#define SL 1024
#define HD 64
#define NP 513
#define PPAD 520        // scoreR row stride in workspace
#define BK 544          // bucket contraction padded to 8 waves * 68
#define SROW (SL + 4)   // LDS score row stride (bank-conflict padding)

typedef __attribute__((ext_vector_type(2))) float v2f;
typedef __attribute__((ext_vector_type(8))) float v8f;

// ---------------------------------------------------------------------------
// Kernel 1: scoreR[b,i,p] = dot(Q[b,0,i,:], relation[b,p,:])   (WMMA f32 GEMM)
// ---------------------------------------------------------------------------
__global__ __launch_bounds__(32) void raa_relscore_kernel(
    const float* __restrict__ q, const float* __restrict__ rel,
    float* __restrict__ scoreR) {
  const int lane = threadIdx.x;
  const int hf   = lane >> 4;      // half-wave select
  const int m    = lane & 15;
  const int off  = hf * 2;         // K sub-offset per fp32 WMMA layout

  const int b  = blockIdx.z;
  const int i0 = blockIdx.y * 16;
  const int p0 = blockIdx.x * 16;

  const int prow  = p0 + m;
  // Clamp instead of predicating: a clamped lane only pollutes output
  // column N==lane, which is never stored.
  const int prow_c = prow < (NP - 1) ? prow : (NP - 1);

  const float* qb = q   + ((size_t)b * NH * SL + i0 + m) * HD + off;  // head 0
  const float* rb = rel + ((size_t)b * NP + prow_c) * HD + off;

  // Preload both fragment sets, then run the WMMA chain (32 loads in flight).
  v2f aq[16], bq[16];
#pragma unroll
  for (int kk = 0; kk < 16; ++kk) aq[kk] = *(const v2f*)(qb + kk * 4);
#pragma unroll
  for (int kk = 0; kk < 16; ++kk) bq[kk] = *(const v2f*)(rb + kk * 4);

  v8f acc = {0.f, 0.f, 0.f, 0.f, 0.f, 0.f, 0.f, 0.f};
#pragma unroll
  for (int kk = 0; kk < 16; ++kk)
    acc = __builtin_amdgcn_wmma_f32_16x16x4_f32(false, aq[kk], false, bq[kk],
                                                (short)0, acc, false, false);

  if (prow < NP) {
#pragma unroll
    for (int r = 0; r < 8; ++r) {
      int row = i0 + r + hf * 8;
      scoreR[((size_t)b * SL + row) * PPAD + prow] = acc[r];
    }
  }
}

// ---------------------------------------------------------------------------
// Kernel 2: full relation-aware attention for one (b,h,16-row tile)
// ---------------------------------------------------------------------------
__global__ __launch_bounds__(256) void raa_attn_kernel(
    const float* __restrict__ q, const float* __restrict__ k,
    const float* __restrict__ v, const float* __restrict__ rel,
    const int* __restrict__ pmap, const int* __restrict__ mask,
    const float* __restrict__ scoreR,
    float* __restrict__ out_attn, float* __restrict__ out_p) {
  __shared__ float sS[16][SROW];      // score strip / normalized probs
  __shared__ float sBk[16][BK];       // relation buckets (zero-padded)
  __shared__ float sOut[16][HD + 4];  // cross-wave output reduction
  __shared__ float sRed[256];
  __shared__ float sRow[16];

  const int tid  = threadIdx.x;
  const int wave = tid >> 5;
  const int lane = tid & 31;
  const int hf   = lane >> 4;
  const int m    = lane & 15;
  const int off  = hf * 2;

  const int bh = blockIdx.y;
  const int b  = bh / NH;
  const int h  = bh % NH;
  const int i0 = blockIdx.x * 16;

  for (int t = tid; t < 16 * BK; t += 256) ((float*)sBk)[t] = 0.f;
  for (int t = tid; t < 16 * (HD + 4); t += 256) ((float*)sOut)[t] = 0.f;

  // Preload Q-tile A fragments (one float2 per lane per K-step).
  const float* qb = q + (((size_t)b * NH + h) * SL + i0 + m) * HD + off;
  v2f aq[16];
#pragma unroll
  for (int kk = 0; kk < 16; ++kk) aq[kk] = *(const v2f*)(qb + kk * 4);

  const float* kb  = k + (((size_t)b * NH + h) * SL) * HD;
  const float* srb = scoreR + ((size_t)b * SL) * PPAD;
  const int*   pmb = pmap + ((size_t)b * SL) * SL;
  const int*   mkb = mask + ((size_t)b * SL) * SL;

  // ---- Phase A: scores = QK^T + gather(scoreR, path_map), scale, mask ----
  for (int jt = wave; jt < SL / 16; jt += 8) {
    const int j0 = jt * 16;
    const float* kt = kb + (size_t)(j0 + m) * HD + off;
    // Batch all 16 B-fragment loads before the WMMA chain.
    v2f bq[16];
#pragma unroll
    for (int kk = 0; kk < 16; ++kk) bq[kk] = *(const v2f*)(kt + kk * 4);
    v8f acc = {0.f, 0.f, 0.f, 0.f, 0.f, 0.f, 0.f, 0.f};
#pragma unroll
    for (int kk = 0; kk < 16; ++kk)
      acc = __builtin_amdgcn_wmma_f32_16x16x4_f32(false, aq[kk], false, bq[kk],
                                                  (short)0, acc, false, false);
#pragma unroll
    for (int r = 0; r < 8; ++r) {
      int rl  = r + hf * 8;
      int row = i0 + rl;
      int col = j0 + m;
      int pidx = pmb[(size_t)row * SL + col];
      float g  = srb[(size_t)row * PPAD + pidx];
      float s  = (acc[r] + g) * 0.125f;  // 1/sqrt(64)
      if (mkb[(size_t)row * SL + col] == 0) s = -1e9f;
      sS[rl][col] = s;
    }
  }
  __syncthreads();

  // ---- Phase B: softmax per row, emit p_attn, scatter into buckets ----
  const int row = tid >> 4;
  const int t16 = tid & 15;
  float pmax = -3.4e38f;
  for (int c = t16; c < SL; c += 16) pmax = fmaxf(pmax, sS[row][c]);
  sRed[tid] = pmax;
  __syncthreads();
  if (t16 == 0) {
    float mx = sRed[tid];
    for (int u = 1; u < 16; ++u) mx = fmaxf(mx, sRed[tid + u]);
    sRow[row] = mx;
  }
  __syncthreads();
  const float mx = sRow[row];
  float ps = 0.f;
  for (int c = t16; c < SL; c += 16) {
    float e = __expf(sS[row][c] - mx);
    sS[row][c] = e;
    ps += e;
  }
  sRed[tid] = ps;
  __syncthreads();
  if (t16 == 0) {
    float sm = 0.f;
    for (int u = 0; u < 16; ++u) sm += sRed[tid + u];
    sRow[row] = 1.0f / sm;
  }
  __syncthreads();
  const float inv = sRow[row];
  float* pout = out_p + ((((size_t)b * NH + h) * SL + i0 + row) * SL);
  for (int c = t16; c < SL; c += 16) {
    float pv = sS[row][c] * inv;
    sS[row][c] = pv;
    pout[c] = pv;
    int pidx = pmb[(size_t)(i0 + row) * SL + c];
    atomicAdd(&sBk[row][pidx], pv);
  }
  __syncthreads();

  // ---- Phase C: out = P @ V + buckets @ relation (WMMA, split over waves) --
  const float* vb = v + (((size_t)b * NH + h) * SL) * HD;
  v8f accn[4];
  const v8f vz = {0.f, 0.f, 0.f, 0.f, 0.f, 0.f, 0.f, 0.f};
#pragma unroll
  for (int n = 0; n < 4; ++n) accn[n] = vz;

  // P @ V : contraction slice j in [wave*128, wave*128+128)
#pragma unroll 2
  for (int ks = 0; ks < 32; ++ks) {
    const int j0 = wave * 128 + ks * 4;
    v2f a = *(const v2f*)&sS[m][j0 + off];
    // Batch the 8 global B loads for this K-step before the 4 WMMAs.
    v2f bmv[4];
#pragma unroll
    for (int n = 0; n < 4; ++n) {
      const int col = n * 16 + m;
      bmv[n].x = vb[(size_t)(j0 + off) * HD + col];
      bmv[n].y = vb[(size_t)(j0 + off + 1) * HD + col];
    }
#pragma unroll
    for (int n = 0; n < 4; ++n)
      accn[n] = __builtin_amdgcn_wmma_f32_16x16x4_f32(false, a, false, bmv[n],
                                                      (short)0, accn[n], false, false);
  }

  // buckets @ relation : contraction slice p in [wave*68, wave*68+68).
  // Out-of-range K rows are clamped: the matching bucket values are exactly
  // 0.0f, and clamped reads are finite, so they contribute 0 to the output.
  const float* relb = rel + ((size_t)b * NP) * HD;
#pragma unroll 2
  for (int ks = 0; ks < 17; ++ks) {
    const int p0 = wave * 68 + ks * 4;
    v2f a = *(const v2f*)&sBk[m][p0 + off];
    const int pr0 = p0 + off;
    const int pc0 = pr0     < (NP - 1) ? pr0     : (NP - 1);
    const int pc1 = pr0 + 1 < (NP - 1) ? pr0 + 1 : (NP - 1);
    v2f bmv[4];
#pragma unroll
    for (int n = 0; n < 4; ++n) {
      const int col = n * 16 + m;
      bmv[n].x = relb[(size_t)pc0 * HD + col];
      bmv[n].y = relb[(size_t)pc1 * HD + col];
    }
#pragma unroll
    for (int n = 0; n < 4; ++n)
      accn[n] = __builtin_amdgcn_wmma_f32_16x16x4_f32(false, a, false, bmv[n],
                                                      (short)0, accn[n], false, false);
  }

  // Cross-wave reduction of the 16x64 output tile.
#pragma unroll
  for (int n = 0; n < 4; ++n) {
#pragma unroll
    for (int r = 0; r < 8; ++r) {
      int rl  = r + hf * 8;
      int col = n * 16 + m;
      atomicAdd(&sOut[rl][col], accn[n][r]);
    }
  }
  __syncthreads();

  float* ob = out_attn + (((size_t)b * NH + h) * SL + i0) * HD;
  for (int t = tid; t < 16 * HD; t += 256) {
    int rl = t / HD, c = t % HD;
    ob[(size_t)rl * HD + c] = sOut[rl][c];
  }
}

// ---------------------------------------------------------------------------
extern "C" void kernel_launch(void* const* d_in, const int* in_sizes, int n_in,
                              void* d_out, int out_size, void* d_ws, size_t ws_size,
                              hipStream_t stream) {
  const float* q    = (const float*)d_in[0];
  const float* k    = (const float*)d_in[1];
  const float* v    = (const float*)d_in[2];
  const float* rel  = (const float*)d_in[3];
  const int*   pmap = (const int*)d_in[4];
  const int*   mask = (const int*)d_in[5];

  float* out_attn = (float*)d_out;                                   // (BS,H,L,D)
  float* out_p    = out_attn + (size_t)BSZ * NH * SL * HD;           // (BS,H,L,L)
  float* scoreR   = (float*)d_ws;  // needs BSZ*SL*PPAD*4 = ~8.5 MB

  dim3 g1((NP + 15) / 16, SL / 16, BSZ);
  raa_relscore_kernel<<<g1, 32, 0, stream>>>(q, rel, scoreR);

  dim3 g2(SL / 16, BSZ * NH, 1);
  raa_attn_kernel<<<g2, 256, 0, stream>>>(q, k, v, rel, pmap, mask, scoreR,
                                          out_attn, out_p);
}